// CombinedJointUNNModel_39067022524908
// MI455X (gfx1250) — compile-verified
//
#include <hip/hip_runtime.h>
#include <stdint.h>

// ---------------- problem dims (from reference) ----------------
#define INPUTS    128
#define INPUTS_T  130              // +0.0, +1.0 constants
#define HIDDEN    256
#define OUTPUTS   32
#define NODES     288              // HIDDEN + OUTPUTS
#define NEURONS   418              // INPUTS_T + NODES
#define NNET      64
#define KCON      32
#define BATCH     1024

// ---------------- phase-1 tiling ----------------
#define P1_TB     64               // batch samples (== threads) per workgroup
#define P1_TILES  (BATCH / P1_TB)  // 16

typedef float        v2f __attribute__((ext_vector_type(2)));
typedef float        v8f __attribute__((ext_vector_type(8)));
typedef unsigned int v4u __attribute__((ext_vector_type(4)));
typedef int          v4i __attribute__((ext_vector_type(4)));
typedef int          v8i __attribute__((ext_vector_type(8)));

// =====================================================================
// Phase 0: pre-gather weights: wstage[n,j,k] = W[conn[n,j,k], j]
// (so phase 1 reads weights with uniform addresses -> SMEM s_load)
// =====================================================================
__global__ __launch_bounds__(256) void unn_gather_w(
    const float* __restrict__ W, const int* __restrict__ conn,
    float* __restrict__ wstage) {
  int e = blockIdx.x * 256 + threadIdx.x;
  if (e < NNET * NODES * KCON) {
    int node  = (e >> 5) % NODES;       // j
    wstage[e] = W[conn[e] * NODES + node];
  }
}

// =====================================================================
// Phase 1: sparse forward. One WG = (network n, 64-sample tile).
// lane-private post[418] rows live in LDS; stride 418 words gives
// lane bank offset 34 mod 64 -> conflict-free ds_load gathers.
// x tile staged with a TDM tensor_load_to_lds (iterate mode:
// 64 iterations of one 128-float row, lds stride 418 words).
//
// ILP structure: occupancy is LDS-capacity-bound (~1 wave/SIMD), so the
// inner loop processes TWO independent nodes (conn bound idx<129+j means
// node j+1 never reads node j) with FOUR partial accumulators each:
// 64 ds_loads in flight, 8 independent fmac chains, stores only at the end.
// =====================================================================
__global__ __launch_bounds__(P1_TB) void unn_forward(
    const float* __restrict__ x, const int* __restrict__ conn,
    const float* __restrict__ wstage, float* __restrict__ outs) {
  __shared__ float post[P1_TB * NEURONS];   // 64*418*4 = 107,008 B
  const int n    = blockIdx.y;
  const int b0   = blockIdx.x * P1_TB;
  const int tid  = threadIdx.x;
  const int base = tid * NEURONS;

  // ---- stage x[b0..b0+63][0..127] -> post[row*418 + 0..127] via TDM ----
  if (tid < 32) {                            // wave 0 issues for whole WG
    unsigned long long ga = (unsigned long long)(uintptr_t)(x + (size_t)b0 * INPUTS);
    unsigned lds = (unsigned)(uintptr_t)(&post[0]);   // low 32b = LDS byte offset
    // D# group0: count=1 | lds_addr | global_addr | type=2
    v4u g0 = { 1u, lds, (unsigned)(ga & 0xffffffffu),
               (unsigned)((ga >> 32) & 0x01ffffffu) | (2u << 30) };
    // D# group1: data_size=4B (code 2), iterate_enable;
    //   tensor_dim0=128, tensor_dim1=64, tile_dim0=128, tile_dim1=1,
    //   tensor_dim0_stride=128, tensor_dim1_stride=128
    v8i g1 = { (2 << 16) | (1 << 19),
               (INPUTS << 16),        // tensor_dim0[15:0] at bits 63:48
               (P1_TB << 16),         // tensor_dim0 hi=0 | tensor_dim1 lo
               (INPUTS << 16),        // tensor_dim1 hi=0 | tile_dim0
               1,                     // tile_dim1=1 | tile_dim2=0
               INPUTS,                // tensor_dim0_stride lo32
               (INPUTS << 16),        // stride0 hi=0 | tensor_dim1_stride lo
               0 };
    // D# group2 (iterate): tensor_dim2=1, lds_inc=418 elems,
    //   global_inc=128 elems, iterate_count=63 -> 64 rows
    v4i g2 = { 1, NEURONS, INPUTS, ((P1_TB - 1) << 16) };
    v4i g3 = { 0, 0, 0, 0 };
    v8i gx = { 0, 0, 0, 0, 0, 0, 0, 0 };   // trailing operand (clang-23 6-arg form)
    __builtin_amdgcn_tensor_load_to_lds(g0, g1, g2, g3, gx, 0);
    __builtin_amdgcn_s_wait_tensorcnt(0);
  }
  post[base + INPUTS]     = 0.0f;    // bias-0 slot
  post[base + INPUTS + 1] = 1.0f;    // bias-1 slot
  __syncthreads();

  const int*   cn = conn   + n * (NODES * KCON);
  const float* wn = wstage + n * (NODES * KCON);

  for (int j = 0; j < NODES; j += 2) {
    const int*   c0 = cn + j * KCON;        // uniform -> s_load
    const float* w0 = wn + j * KCON;        // uniform -> s_load
    const int*   c1 = c0 + KCON;
    const float* w1 = w0 + KCON;

    float a00 = 0.f, a01 = 0.f, a02 = 0.f, a03 = 0.f;   // node j
    float a10 = 0.f, a11 = 0.f, a12 = 0.f, a13 = 0.f;   // node j+1
#pragma unroll
    for (int k = 0; k < KCON; k += 4) {
      a00 = fmaf(w0[k + 0], post[base + c0[k + 0]], a00);
      a01 = fmaf(w0[k + 1], post[base + c0[k + 1]], a01);
      a02 = fmaf(w0[k + 2], post[base + c0[k + 2]], a02);
      a03 = fmaf(w0[k + 3], post[base + c0[k + 3]], a03);
    }
#pragma unroll
    for (int k = 0; k < KCON; k += 4) {
      a10 = fmaf(w1[k + 0], post[base + c1[k + 0]], a10);
      a11 = fmaf(w1[k + 1], post[base + c1[k + 1]], a11);
      a12 = fmaf(w1[k + 2], post[base + c1[k + 2]], a12);
      a13 = fmaf(w1[k + 3], post[base + c1[k + 3]], a13);
    }
    const float s0 = (a00 + a01) + (a02 + a03);
    const float s1 = (a10 + a11) + (a12 + a13);
    // stores LAST: all 64 gathers above are load-only w.r.t. post
    post[base + INPUTS_T + j]     = (s0 > 0.0f) ? s0 : 0.01f * s0;
    post[base + INPUTS_T + j + 1] = (s1 > 0.0f) ? s1 : 0.01f * s1;
  }

  // store per-network outputs in [n][o][b] order (coalesced, feeds WMMA B)
  const int b = b0 + tid;
#pragma unroll
  for (int o = 0; o < OUTPUTS; ++o)
    outs[(size_t)(n * OUTPUTS + o) * BATCH + b] =
        post[base + INPUTS_T + HIDDEN + o];
}

// =====================================================================
// Phase 2: out[b,o] = sum_n softmax(nw)[n] * outs[n,o,b]
// Exact fp32 WMMA (v_wmma_f32_16x16x4_f32): A = wsoft replicated over M,
// B[k][col] = outs[kb+k, o, b0+col]; 16 chained WMMAs reduce all 64 nets.
// One wave per (o, 16-sample tile); row M=0 of D is the answer.
// =====================================================================
__global__ __launch_bounds__(128) void unn_mix_wmma(
    const float* __restrict__ nw, const float* __restrict__ outs,
    float* __restrict__ out) {
  __shared__ float wsoft[NNET];
  const int tid  = threadIdx.x;
  const int lane = tid & 31;
  const int wave = tid >> 5;           // wave32
  const int o    = blockIdx.y * 4 + wave;
  const int b0   = blockIdx.x * 16;

  // stable softmax over 64 weights (redundant per thread; trivial cost)
  float mx = -3.4e38f;
  for (int i = 0; i < NNET; ++i) mx = fmaxf(mx, nw[i]);
  float s = 0.0f;
  for (int i = 0; i < NNET; ++i) s += __expf(nw[i] - mx);
  float inv = 1.0f / s;
  if (tid < NNET) wsoft[tid] = __expf(nw[tid] - mx) * inv;
  __syncthreads();

  const int half = lane >> 4;          // K half-split (A layout per ISA)
  const int col  = lane & 15;          // N = batch column
  v8f c = {};
#pragma unroll
  for (int kb = 0; kb < NNET; kb += 4) {
    const int k0 = kb + 2 * half;      // this lane-half's K pair
    v2f a, bm;
    a.x  = wsoft[k0];                  // A[m][K0/K2] (replicated over m)
    a.y  = wsoft[k0 + 1];              // A[m][K1/K3]
    bm.x = outs[(size_t)( k0      * OUTPUTS + o) * BATCH + b0 + col];
    bm.y = outs[(size_t)((k0 + 1) * OUTPUTS + o) * BATCH + b0 + col];
    c = __builtin_amdgcn_wmma_f32_16x16x4_f32(false, a, false, bm,
                                              (short)0, c, false, false);
  }
  // D row M=0 lives in VGPR0 lanes 0..15 (all M rows identical here)
  if (lane < 16) out[(size_t)(b0 + col) * OUTPUTS + o] = c[0];
}

// =====================================================================
// Host-side launcher. Workspace use: wstage 589,824 f32 + outs 2,097,152 f32
// (~10.75 MB total).
// =====================================================================
extern "C" void kernel_launch(void* const* d_in, const int* in_sizes, int n_in,
                              void* d_out, int out_size, void* d_ws, size_t ws_size,
                              hipStream_t stream) {
  (void)in_sizes; (void)n_in; (void)out_size; (void)ws_size;
  const float* x    = (const float*)d_in[0];
  const float* W    = (const float*)d_in[1];
  const float* nw   = (const float*)d_in[2];
  const int*   conn = (const int*)d_in[3];
  float* out = (float*)d_out;

  float* wstage = (float*)d_ws;                       // [64][288][32]
  float* outsws = wstage + NNET * NODES * KCON;       // [64][32][1024]

  unn_gather_w<<<(NNET * NODES * KCON + 255) / 256, 256, 0, stream>>>(
      W, conn, wstage);
  unn_forward<<<dim3(P1_TILES, NNET), P1_TB, 0, stream>>>(
      x, conn, wstage, outsws);
  unn_mix_wmma<<<dim3(BATCH / 16, OUTPUTS / 4), 128, 0, stream>>>(
      nw, outsws, out);
}